// QuantumNet_31825707663367
// MI455X (gfx1250) — compile-verified
//
#include <hip/hip_runtime.h>
#include <hip/hip_bf16.h>
#include <math.h>

// Closed form of the reference circuit:
//   E_q = cos(t1_q) + i*sigma_q*sin(t1_q)*cos(t0_q),  sigma_q = +-1 from bit_q
//   prob1[b] = (1 + Re prod_q E_q)/2
//            = (1 + R * cos(S0 - 2*dot(bits[b,:], phi))) / 2
//   phi_q = atan2(sin t1 * cos t0, cos t1), r_q = |E_q| (sigma-independent),
//   R = prod r_q, S0 = sum phi_q, bits = (X > 0.5)
// Dense math left: bits[256x15] @ phi[15] -> V_WMMA_F32_16X16X4_F32 (4 K-steps).
// Per-qubit trig is computed one-qubit-per-lane, combined with wave32
// butterfly reductions; B-fragment phi values gathered with lane permutes.

typedef __attribute__((ext_vector_type(2))) float v2f;
typedef __attribute__((ext_vector_type(8))) float v8f;

#define DQ 15

__global__ __launch_bounds__(32)
void qnet_wmma_kernel(const float* __restrict__ X,
                      const float* __restrict__ P,
                      float* __restrict__ out)
{
    const int lane = threadIdx.x & 31;
    const int tile = blockIdx.x;          // 16 batch rows per tile

    // --- per-qubit coefficients: lane q handles qubit q (q < 15) ---
    const bool valid = lane < DQ;
    const int  q     = valid ? lane : 0;
    const float t0 = P[q];        // params[0][q]  (XX layer)
    const float t1 = P[DQ + q];   // params[1][q]  (ZZ layer)
    const float c  = __cosf(t1);
    const float d  = __sinf(t1) * __cosf(t0);
    float r  = valid ? __fsqrt_rn(c * c + d * d) : 1.0f;
    float ph = valid ? atan2f(d, c)              : 0.0f;

    // wave32 butterfly reductions: R = prod r, S0 = sum phi (identities in lanes >= 15)
    float R = r, S0 = ph;
#pragma unroll
    for (int m = 16; m >= 1; m >>= 1) {
        R  *= __shfl_xor(R,  m, 32);
        S0 += __shfl_xor(S0, m, 32);
    }

    // --- WMMA fragments ---
    // A (16x4 f32): lane<16 -> row M=lane, regs hold K {0,1}; lane>=16 -> same row, K {2,3}
    // B (4x16 f32): value depends only on K (phi broadcast over N), mirrored k = 2*half + reg
    const int row   = lane & 15;
    const int khalf = lane >> 4;
    const float* Xrow = X + (tile * 16 + row) * DQ;

    v8f acc = {0.f, 0.f, 0.f, 0.f, 0.f, 0.f, 0.f, 0.f};
#pragma unroll
    for (int kc = 0; kc < 3; ++kc) {
        const int kb = kc * 4 + khalf * 2;      // compile-time offset + khalf select
        v2f a, b;
        a.x = (Xrow[kb]     > 0.5f) ? 1.0f : 0.0f;
        a.y = (Xrow[kb + 1] > 0.5f) ? 1.0f : 0.0f;
        b.x = __shfl(ph, kb,     32);           // per-lane index -> ds_bpermute
        b.y = __shfl(ph, kb + 1, 32);
        acc = __builtin_amdgcn_wmma_f32_16x16x4_f32(
            false, a, false, b, (short)0, acc, false, false);
    }
    {   // last K-chunk: K = 12..15, with K=15 padded (phi lane15 already 0)
        const int kb = 12 + khalf * 2;
        v2f a, b;
        const float x13 = Xrow[13];             // valid for both halves
        a.x = (Xrow[kb] > 0.5f) ? 1.0f : 0.0f;
        a.y = (khalf == 0 && x13 > 0.5f) ? 1.0f : 0.0f;  // K=15 -> 0
        b.x = __shfl(ph, kb,     32);
        b.y = __shfl(ph, kb + 1, 32);           // kb+1==15 -> ph=0
        acc = __builtin_amdgcn_wmma_f32_16x16x4_f32(
            false, a, false, b, (short)0, acc, false, false);
    }

    // D layout (16x16 f32): lane l<16 holds column N=l, rows v in acc[v];
    // lane l>=16 holds rows v+8. All columns equal the per-row dot product.
    if (lane == 0) {
#pragma unroll
        for (int v = 0; v < 8; ++v)
            out[tile * 16 + v] = 0.5f * (1.0f + R * __cosf(S0 - 2.0f * acc[v]));
    } else if (lane == 16) {
#pragma unroll
        for (int v = 0; v < 8; ++v)
            out[tile * 16 + 8 + v] = 0.5f * (1.0f + R * __cosf(S0 - 2.0f * acc[v]));
    }
}

extern "C" void kernel_launch(void* const* d_in, const int* in_sizes, int n_in,
                              void* d_out, int out_size, void* d_ws, size_t ws_size,
                              hipStream_t stream) {
    const float* X = (const float*)d_in[0];   // [256, 15] float32
    const float* P = (const float*)d_in[1];   // [2, 15] float32
    float* out = (float*)d_out;               // [256] float32

    const int B = in_sizes[0] / DQ;           // 256
    const int tiles = B / 16;                 // 16 tiles of 16 rows

    qnet_wmma_kernel<<<tiles, 32, 0, stream>>>(X, P, out);
}